// Model_41266045780592
// MI455X (gfx1250) — compile-verified
//
#include <hip/hip_runtime.h>
#include <hip/hip_bf16.h>
#include <math.h>

typedef float v2f __attribute__((ext_vector_type(2)));
typedef float v4f __attribute__((ext_vector_type(4)));
typedef float v8f __attribute__((ext_vector_type(8)));

#define HIDDEN 32
#define CCH    256
#define BB     4
#define DD     32
#define HH2    64
#define WW2    64
#define SLICE  (DD * HH2 * WW2)      /* 131072 = 2^17 */
#define PADVOL (36 * 68 * 68)        /* 166464 */

// ---------------------------------------------------------------------------
// Kernel 1: weighted per-(b,c) mean equivalent to mean over the reflect-padded
// volume. Reflect pad=2 double-counts indices {1,2,N-3,N-2} per dim.
// ---------------------------------------------------------------------------
__device__ __forceinline__ float wgt(int i, int n) {
  return (i == 1 || i == 2 || i == n - 3 || i == n - 2) ? 2.f : 1.f;
}

__global__ void __launch_bounds__(256) k_wmean(const float* __restrict__ x,
                                               float* __restrict__ means) {
  const int bc = blockIdx.x;                       // b*256 + c
  const v4f* xs = (const v4f*)(x + (size_t)bc * SLICE);
  float sum = 0.f;
  for (int i = threadIdx.x; i < SLICE / 4; i += 256) {
    const int e  = i << 2;
    const int d  = e >> 12;          // /(64*64)
    const int h  = (e >> 6) & 63;
    const int w0 = e & 63;           // 4-aligned, no wrap inside a v4f
    const float wdh = wgt(d, 32) * wgt(h, 64);
    const v4f v = xs[i];
    float s = v.x * wgt(w0, 64) + v.y * wgt(w0 + 1, 64) +
              v.z * wgt(w0 + 2, 64) + v.w * wgt(w0 + 3, 64);
    sum = fmaf(wdh, s, sum);
  }
  __shared__ float red[256];
  red[threadIdx.x] = sum;
  __syncthreads();
  for (int off = 128; off > 0; off >>= 1) {
    if (threadIdx.x < off) red[threadIdx.x] += red[threadIdx.x + off];
    __syncthreads();
  }
  if (threadIdx.x == 0) means[bc] = red[0] * (1.f / (float)PADVOL);
}

// ---------------------------------------------------------------------------
// Kernel 2: single-wave LSTM scan (256 steps), recurrent GEMM on f32 WMMA.
//   z(4x128) = h(4x32) @ W_hh^T(32x128) + inp*W_ih + bias
//   A = h padded to 16x32 (8 K-slices), B = W_hh^T stationary in VGPRs.
//   h lives in a 16-row LDS buffer (rows 4..15 permanently zero) so the
//   per-step A build is unconditional ds_loads — no exec-mask branches.
// ---------------------------------------------------------------------------
__device__ __forceinline__ float sigf(float x) { return 1.f / (1.f + __expf(-x)); }
__device__ __forceinline__ float tanh_f(float x) {
  float ax = fminf(fabsf(x), 15.f);
  float e  = __expf(2.f * ax);
  return copysignf((e - 1.f) / (e + 1.f), x);
}

__global__ void __launch_bounds__(32) k_lstm(
    const float* __restrict__ W_ih, const float* __restrict__ W_hh,
    const float* __restrict__ b_ih, const float* __restrict__ b_hh,
    const float* __restrict__ gate_w, const float* __restrict__ gate_b,
    const float* __restrict__ means, float* __restrict__ gates) {
  __shared__ float h_lds[16 * HIDDEN];   // padded h: rows 0..3 = batch, 4..15 = 0
  __shared__ float seq[BB * CCH];        // channel means
  const int lane  = threadIdx.x;         // 0..31 (wave32)
  const int nl    = lane & 15;
  const int khalf = (lane >> 4) << 1;    // 0 for lanes 0-15, 2 for 16-31

  for (int i = lane; i < BB * CCH; i += 32) seq[i] = means[i];
  for (int i = lane; i < 16 * HIDDEN; i += 32) h_lds[i] = 0.f;

  // B tiles: B[t][s] is the 4x16 slice  W_hh^T[4s..4s+3][16t..16t+15].
  // WMMA-B f32 4x16 layout: v.x -> K = 4s+khalf, v.y -> K = 4s+khalf+1, N = nl.
  v2f Bt[8][8];
  for (int t = 0; t < 8; ++t)
    for (int s = 0; s < 8; ++s) {
      const int row = t * 16 + nl;        // gate-output index (N)
      const int col = s * 4 + khalf;      // hidden index (K)
      Bt[t][s].x = W_hh[row * HIDDEN + col];
      Bt[t][s].y = W_hh[row * HIDDEN + col + 1];
    }
  float biasv[8], wihv[8];
  for (int t = 0; t < 8; ++t) {
    const int cidx = t * 16 + nl;
    biasv[t] = b_ih[cidx] + b_hh[cidx];
    wihv[t]  = W_ih[cidx];                // input_size == 1
  }
  const float gwL = gate_w[nl], gwH = gate_w[nl + 16];
  const float gb  = gate_b[0];

  float cst[8];                            // c[b][j], j in {nl, nl+16}
#pragma unroll
  for (int k = 0; k < 8; ++k) cst[k] = 0.f;

  __syncthreads();

  for (int t = 0; t < CCH; ++t) {
    // Build A (16x32 padded h): lane row M = nl; rows >= 4 read hard zeros.
    v2f A[8];
#pragma unroll
    for (int s = 0; s < 8; ++s) {
      const int k0 = s * 4 + khalf;
      A[s].x = h_lds[nl * HIDDEN + k0];
      A[s].y = h_lds[nl * HIDDEN + k0 + 1];
    }
    float inp[4];
#pragma unroll
    for (int b = 0; b < 4; ++b) inp[b] = seq[b * CCH + t];

    __syncthreads();  // everyone captured h into A before we overwrite h_lds

    float gl[4] = {0.f, 0.f, 0.f, 0.f};
    // grp 0 -> tiles {0,2,4,6}: hidden j = nl ; grp 1 -> {1,3,5,7}: j = nl+16
#pragma unroll
    for (int grp = 0; grp < 2; ++grp) {
      v8f D[4];                            // q: 0=i 1=f 2=g 3=o
#pragma unroll
      for (int q = 0; q < 4; ++q) {
        const int ti = q * 2 + grp;
        v8f c;
#pragma unroll
        for (int r = 0; r < 8; ++r) c[r] = fmaf(inp[r & 3], wihv[ti], biasv[ti]);
#pragma unroll
        for (int s = 0; s < 8; ++s)
          c = __builtin_amdgcn_wmma_f32_16x16x4_f32(false, A[s], false, Bt[ti][s],
                                                    (short)0, c, false, false);
        D[q] = c;
      }
      // Activations: lane nl holds z[b][col] at D[q][b] for b = 0..3.
#pragma unroll
      for (int b = 0; b < 4; ++b) {
        float& cc = cst[b * 2 + grp];
        cc = sigf(D[1][b]) * cc + sigf(D[0][b]) * tanh_f(D[2][b]);
        const float hv = sigf(D[3][b]) * tanh_f(cc);
        if (lane < 16) h_lds[b * HIDDEN + nl + grp * 16] = hv;
        gl[b] = fmaf(hv, grp ? gwH : gwL, gl[b]);
      }
    }
    // Reduce gate logits over lanes 0..15, emit sigmoid gate for this step.
#pragma unroll
    for (int off = 8; off > 0; off >>= 1)
#pragma unroll
      for (int b = 0; b < 4; ++b) gl[b] += __shfl_down(gl[b], off, 16);
    if (lane == 0)
#pragma unroll
      for (int b = 0; b < 4; ++b) gates[b * CCH + t] = sigf(gl[b] + gb);
    __syncthreads();  // new h visible before next step's A build
  }
}

// ---------------------------------------------------------------------------
// Kernel 3: out = x * gate[b][c].  x is 512MB (2^27 floats);
// output stores are non-temporal so they don't evict x.
// ---------------------------------------------------------------------------
__global__ void __launch_bounds__(256) k_scale(const v4f* __restrict__ x4,
                                               const float* __restrict__ gates,
                                               v4f* __restrict__ out4) {
  const size_t i = (size_t)blockIdx.x * 256 + threadIdx.x;  // float4 index
  const int bc   = (int)(i >> 15);        // (i*4) >> 17, SLICE = 2^17
  const float g  = gates[bc];
  v4f v = x4[i];
  v.x *= g; v.y *= g; v.z *= g; v.w *= g;
  __builtin_nontemporal_store(v, &out4[i]);
}

// ---------------------------------------------------------------------------
extern "C" void kernel_launch(void* const* d_in, const int* in_sizes, int n_in,
                              void* d_out, int out_size, void* d_ws, size_t ws_size,
                              hipStream_t stream) {
  const float* x      = (const float*)d_in[0];
  const float* W_ih   = (const float*)d_in[1];
  const float* W_hh   = (const float*)d_in[2];
  const float* b_ih   = (const float*)d_in[3];
  const float* b_hh   = (const float*)d_in[4];
  const float* gate_w = (const float*)d_in[5];
  const float* gate_b = (const float*)d_in[6];
  float* out   = (float*)d_out;
  float* means = (float*)d_ws;           // 1024 floats
  float* gates = means + BB * CCH;       // 1024 floats

  k_wmean<<<BB * CCH, 256, 0, stream>>>(x, means);
  k_lstm<<<1, 32, 0, stream>>>(W_ih, W_hh, b_ih, b_hh, gate_w, gate_b, means, gates);
  const int n4 = (BB * CCH * SLICE) / 4;           // 33554432 float4s
  k_scale<<<n4 / 256, 256, 0, stream>>>((const v4f*)x, gates, (v4f*)out);
}